// NMTModel_21165598835036
// MI455X (gfx1250) — compile-verified
//
#include <hip/hip_runtime.h>
#include <hip/hip_bf16.h>

// Problem constants from setup_inputs(): B=32, N=64, L=128, D=256
#define BB 32
#define NN 64
#define LL 128
#define DD 256
#define NBATCH (BB * NN)      // 2048
#define MAXLEN (NN * LL)      // 8192
#define ROWS 16               // t-rows per gather block
#define TSPLIT 16             // t-dimension split for pooling
#define BD (BB * DD)          // row stride (in floats) of mb_out: 8192

typedef float v2f __attribute__((ext_vector_type(2)));
typedef float v8f __attribute__((ext_vector_type(8)));

// D = A(16x4) x B(4x16) + C, all f32. Confirmed to lower to
// v_wmma_f32_16x16x4_f32 on gfx1250.
__device__ __forceinline__ v8f wmma_acc(v2f a, v2f b, v8f c) {
    return __builtin_amdgcn_wmma_f32_16x16x4_f32(
        /*neg_a=*/false, a, /*neg_b=*/false, b,
        /*c_mod=*/(short)0, c, /*reuse_a=*/false, /*reuse_b=*/false);
}

// ---------------------------------------------------------------------------
// Kernel 1: per-batch exclusive prefix sum of node lengths (N=64 per batch),
// totals, and lengths_out (float). Tiny: one wave.
// ---------------------------------------------------------------------------
__global__ void k_setup(const int* __restrict__ lengths,
                        float* __restrict__ len_out,
                        int* __restrict__ off,
                        int* __restrict__ tot) {
    const int b = threadIdx.x;
    if (b < BB) {
        int s = 0;
        for (int n = 0; n < NN; ++n) {
            off[b * NN + n] = s;
            s += lengths[b * NN + n];
        }
        tot[b] = s;
        len_out[b] = (float)s;
    }
}

// ---------------------------------------------------------------------------
// Kernel 2: gather-formulated reconstruction. Each output slot (t, b) is
// written exactly once: binary-search node offsets (in LDS) to map t ->
// (node, pos); copy the D=256 row with float4 (B128), or write the pad value.
// This is the bandwidth-dominant phase (~268 MB read + ~268 MB write).
// ---------------------------------------------------------------------------
__global__ void k_gather(const float* __restrict__ src,
                         const float* __restrict__ mb,
                         const int* __restrict__ recover,
                         const int* __restrict__ off,
                         const int* __restrict__ tot,
                         float* __restrict__ src_out,
                         float* __restrict__ mb_out) {
    __shared__ int off_s[NN];
    __shared__ int tot_s;
    const int b      = blockIdx.y;
    const int t_base = blockIdx.x * ROWS;
    const int tid    = threadIdx.x;           // 256 threads
    if (tid < NN) off_s[tid] = off[b * NN + tid];
    if (tid == 0) tot_s = tot[b];
    __syncthreads();

    const int sub  = tid >> 6;                // 4 rows in flight
    const int lane = tid & 63;                // 64 lanes * float4 = 256 floats

    for (int r = sub; r < ROWS; r += 4) {
        const int t = t_base + r;
        float4 v = make_float4(0.f, 0.f, 0.f, 0.f);   // pad value for mb
        float sv = 1.0f;                              // pad value for src
        if (t < tot_s) {
            // largest n with off_s[n] <= t  (offsets are nondecreasing)
            int lo = 0, hi = NN - 1;
            while (lo < hi) {
                const int mid = (lo + hi + 1) >> 1;
                if (off_s[mid] <= t) lo = mid; else hi = mid - 1;
            }
            const int pos  = t - off_s[lo];
            const int node = b * NN + lo;
            const float* row = mb + ((size_t)pos * NBATCH + node) * DD;
            v = *(const float4*)(row + lane * 4);
            if (lane == 0) sv = src[(size_t)pos * NBATCH + recover[node]];
        }
        *(float4*)(mb_out + ((size_t)t * BB + b) * DD + lane * 4) = v;
        if (lane == 0) src_out[(size_t)t * BB + b] = sv;
    }
}

// ---------------------------------------------------------------------------
// Kernel 3: mean-pool reduction via the f32 matrix pipe.
//   H[b,d] = sum_{t<tot[b]} mb_out[t,b,d]
// A = all-ones 16x4; B = (4 t-rows x 16 d-cols) tile of mb_out, so every row
// of D accumulates the column sums; row M=0 (C VGPR0, lanes 0..15) is kept.
// One wave per block -> EXEC all-ones (WMMA requirement).
// Main loop is unguarded (full 8-row chunks, 2 independent accumulators for
// ILP, loads batched ahead of the wmmas); only the ragged tail is guarded.
// Partials per t-split go to workspace (no float atomics -> deterministic).
// ---------------------------------------------------------------------------
__global__ void k_pool(const float* __restrict__ mb_out,
                       const int* __restrict__ tot,
                       float* __restrict__ part) {
    const int dg = blockIdx.x;                // 16-wide d-column group (16 groups)
    const int b  = blockIdx.y;
    const int ts = blockIdx.z;                // t-split index
    const int T  = tot[b];
    // rows per split, multiple of 32
    const int per = ((T + TSPLIT * 32 - 1) / (TSPLIT * 32)) * 32;
    const int t0  = ts * per;
    const int t1  = min(T, t0 + per);

    const int lane = threadIdx.x;             // 0..31, single wave
    const int n    = lane & 15;               // column within d-group
    const int h2   = (lane >> 4) << 1;        // 2*h: first K row this lane holds
    const float* base = mb_out + (size_t)b * DD + (size_t)dg * 16 + n;

    v2f a; a.x = 1.0f; a.y = 1.0f;            // ones A-matrix (16x4)
    v8f c0 = {}, c1 = {};

    int t = t0;
    const int tEndMain = t0 + ((t1 - t0) & ~7);   // full 8-row chunks
    for (; t < tEndMain; t += 8) {
        // 4 unguarded strided loads, issued together
        const float b00 = base[(size_t)(t + h2)     * BD];
        const float b01 = base[(size_t)(t + h2 + 1) * BD];
        const float b10 = base[(size_t)(t + 4 + h2)     * BD];
        const float b11 = base[(size_t)(t + 4 + h2 + 1) * BD];
        __builtin_prefetch(base + (size_t)min(t + 16, t1 - 1) * BD, 0, 1);
        v2f bv0; bv0.x = b00; bv0.y = b01;
        v2f bv1; bv1.x = b10; bv1.y = b11;
        c0 = wmma_acc(a, bv0, c0);
        c1 = wmma_acc(a, bv1, c1);
    }
    for (; t < t1; t += 4) {                  // ragged tail (last split only)
        const int k0 = t + h2;
        const int k1 = k0 + 1;
        v2f bv;
        bv.x = (k0 < t1) ? base[(size_t)k0 * BD] : 0.0f;
        bv.y = (k1 < t1) ? base[(size_t)k1 * BD] : 0.0f;
        c0 = wmma_acc(a, bv, c0);
    }

    // C layout: VGPR0, lanes 0..15 hold D[M=0, N=lane]
    const float r = c0[0] + c1[0];
    if (lane < 16) part[((size_t)ts * BB + b) * DD + dg * 16 + lane] = r;
}

// ---------------------------------------------------------------------------
// Kernel 4: deterministic sum of t-split partials, divide by length.
// ---------------------------------------------------------------------------
__global__ void k_finalize(const float* __restrict__ part,
                           const int* __restrict__ tot,
                           float* __restrict__ eh) {
    const int i = blockIdx.x * blockDim.x + threadIdx.x;  // over B*D
    if (i >= BB * DD) return;
    const int b = i / DD;
    float s = 0.f;
    for (int ts = 0; ts < TSPLIT; ++ts) s += part[(size_t)ts * BB * DD + i];
    eh[i] = s / (float)tot[b];
}

// ---------------------------------------------------------------------------
extern "C" void kernel_launch(void* const* d_in, const int* in_sizes, int n_in,
                              void* d_out, int out_size, void* d_ws, size_t ws_size,
                              hipStream_t stream) {
    const float* src     = (const float*)d_in[0];   // [L, NB, 1]
    const float* mb      = (const float*)d_in[1];   // [L, NB, D]
    const int*   lengths = (const int*)d_in[2];     // [NB]
    const int*   recover = (const int*)d_in[3];     // [NB]
    (void)in_sizes; (void)n_in; (void)out_size; (void)ws_size;

    float* out      = (float*)d_out;
    float* src_out  = out;                                          // MAXLEN*B
    float* mb_out   = out + (size_t)MAXLEN * BB;                    // MAXLEN*B*D
    float* eh_out   = mb_out + (size_t)MAXLEN * BB * DD;            // B*D
    float* len_out  = eh_out + (size_t)BB * DD;                     // B

    int*   off_ws  = (int*)d_ws;                    // B*N ints
    int*   tot_ws  = off_ws + NBATCH;               // B ints
    float* part_ws = (float*)(tot_ws + BB);         // TSPLIT*B*D floats

    k_setup<<<dim3(1), dim3(32), 0, stream>>>(lengths, len_out, off_ws, tot_ws);

    dim3 g2(MAXLEN / ROWS, BB);
    k_gather<<<g2, dim3(256), 0, stream>>>(src, mb, recover, off_ws, tot_ws,
                                           src_out, mb_out);

    dim3 g3(DD / 16, BB, TSPLIT);
    k_pool<<<g3, dim3(32), 0, stream>>>(mb_out, tot_ws, part_ws);

    k_finalize<<<dim3((BB * DD) / 256), dim3(256), 0, stream>>>(part_ws, tot_ws, eh_out);
}